// superpixel_loss_13408887898282
// MI455X (gfx1250) — compile-verified
//
#include <hip/hip_runtime.h>
#include <stdint.h>

#define BATCH 8
#define CH 4
#define HW (1024 * 1024)
#define NUM_LABELS 1025
#define LINE_THRESH (255.0f / 4.0f)

#define BLOCKS1_PER_IMG 64
#define BLOCKS2_PER_IMG 128
#define TPB 256

#if __has_builtin(__builtin_amdgcn_global_load_async_to_lds_b32) && \
    __has_builtin(__builtin_amdgcn_s_wait_asynccnt)
#define HAVE_ASYNC_LDS 1
#else
#define HAVE_ASYNC_LDS 0
#endif

typedef __attribute__((address_space(1))) int* gint_p;
typedef __attribute__((address_space(3))) int* lint_p;
typedef float __attribute__((ext_vector_type(4))) float4v;
typedef int __attribute__((ext_vector_type(4))) int4v;

// ---------------------------------------------------------------------------
// ws layout (floats): [sums  B*1025*4][cnts  B*1025][means B*1025*4]
// ---------------------------------------------------------------------------

__global__ __launch_bounds__(TPB) void init_kernel(float* __restrict__ p, int n,
                                                   float* __restrict__ out) {
  int i = blockIdx.x * TPB + threadIdx.x;
  if (i < n) p[i] = 0.0f;
  if (i == 0) out[0] = 0.0f;
}

// Pass 1: per-image LDS-privatized segment sums (ds_add_f32), B128 streaming loads,
// flush via global atomics.
__global__ __launch_bounds__(TPB) void pass1_segsum(const float* __restrict__ Is,
                                                    const int* __restrict__ Ispp,
                                                    float* __restrict__ sums,
                                                    float* __restrict__ cnts) {
  __shared__ float s_sum[NUM_LABELS * CH];  // 16.4 KB
  __shared__ float s_cnt[NUM_LABELS];       //  4.1 KB
  const int tid = threadIdx.x;
  for (int i = tid; i < NUM_LABELS * CH; i += TPB) s_sum[i] = 0.0f;
  for (int i = tid; i < NUM_LABELS; i += TPB) s_cnt[i] = 0.0f;
  __syncthreads();

  const int b = blockIdx.x / BLOCKS1_PER_IMG;
  const int t = blockIdx.x % BLOCKS1_PER_IMG;
  const int pixPerBlk = HW / BLOCKS1_PER_IMG;  // 16384, multiple of TPB*4
  const int base = t * pixPerBlk;

  const float* __restrict__ isb = Is + (size_t)b * CH * HW;
  const int* __restrict__ lb = Ispp + (size_t)b * HW;

  for (int p = base + tid * 4; p < base + pixPerBlk; p += TPB * 4) {
    int4v lab4 = *(const int4v*)(lb + p);
    float4v c0 = *(const float4v*)(isb + p);
    float4v c1 = *(const float4v*)(isb + p + HW);
    float4v c2 = *(const float4v*)(isb + p + 2 * HW);
    float4v c3 = *(const float4v*)(isb + p + 3 * HW);
#pragma unroll
    for (int k = 0; k < 4; ++k) {
      int lab = lab4[k];
      atomicAdd(&s_sum[lab * CH + 0], c0[k]);
      atomicAdd(&s_sum[lab * CH + 1], c1[k]);
      atomicAdd(&s_sum[lab * CH + 2], c2[k]);
      atomicAdd(&s_sum[lab * CH + 3], c3[k]);
      atomicAdd(&s_cnt[lab], 1.0f);
    }
  }
  __syncthreads();

  float* __restrict__ gsum = sums + (size_t)b * NUM_LABELS * CH;
  float* __restrict__ gcnt = cnts + (size_t)b * NUM_LABELS;
  for (int l = tid; l < NUM_LABELS; l += TPB) {
    float c = s_cnt[l];
    if (c != 0.0f) {
      atomicAdd(&gcnt[l], c);
      atomicAdd(&gsum[l * CH + 0], s_sum[l * CH + 0]);
      atomicAdd(&gsum[l * CH + 1], s_sum[l * CH + 1]);
      atomicAdd(&gsum[l * CH + 2], s_sum[l * CH + 2]);
      atomicAdd(&gsum[l * CH + 3], s_sum[l * CH + 3]);
    }
  }
}

// Pass 1.5: means = sums / max(cnt, 1)
__global__ __launch_bounds__(TPB) void means_kernel(const float* __restrict__ sums,
                                                    const float* __restrict__ cnts,
                                                    float* __restrict__ means) {
  int i = blockIdx.x * TPB + threadIdx.x;
  if (i < BATCH * NUM_LABELS) {
    float inv = 1.0f / fmaxf(cnts[i], 1.0f);
    float4v m;
    m.x = sums[i * CH + 0] * inv;
    m.y = sums[i * CH + 1] * inv;
    m.z = sums[i * CH + 2] * inv;
    m.w = sums[i * CH + 3] * inv;
    *(float4v*)(means + (size_t)i * CH) = m;
  }
}

// Pass 2: async-copy the 16.4KB per-image mean table into LDS, stream pixels with
// B128 loads, wave32 shuffle reduction, one scaled atomicAdd per block.
__global__ __launch_bounds__(TPB) void pass2_loss(const float* __restrict__ Is,
                                                  const int* __restrict__ Ispp,
                                                  const float* __restrict__ Il,
                                                  const float* __restrict__ means,
                                                  float* __restrict__ out) {
  __shared__ float s_mean[NUM_LABELS * CH];  // 16.4 KB
  __shared__ float s_red[TPB / 32];
  const int tid = threadIdx.x;
  const int b = blockIdx.x / BLOCKS2_PER_IMG;
  const int t = blockIdx.x % BLOCKS2_PER_IMG;

  const float* __restrict__ mb = means + (size_t)b * NUM_LABELS * CH;

#if HAVE_ASYNC_LDS
  for (int i = tid; i < NUM_LABELS * CH; i += TPB) {
    float* gp = const_cast<float*>(mb + i);
    __builtin_amdgcn_global_load_async_to_lds_b32(
        (gint_p)gp,
        (lint_p)&s_mean[i],
        /*offset=*/0, /*cpol=*/0);
  }
  __builtin_amdgcn_s_wait_asynccnt(0);
#else
  for (int i = tid; i < NUM_LABELS * CH; i += TPB) s_mean[i] = mb[i];
#endif
  __syncthreads();

  const int pixPerBlk = HW / BLOCKS2_PER_IMG;  // 8192, multiple of TPB*4
  const int base = t * pixPerBlk;
  const float* __restrict__ isb = Is + (size_t)b * CH * HW;
  const int* __restrict__ lb = Ispp + (size_t)b * HW;
  const float* __restrict__ ilb = Il + (size_t)b * HW;

  float acc = 0.0f;
  for (int p = base + tid * 4; p < base + pixPerBlk; p += TPB * 4) {
    int4v lab4 = *(const int4v*)(lb + p);
    // Il is single-use: nontemporal so it doesn't evict the L2-resident Is/Ispp set.
    float4v il4 = __builtin_nontemporal_load((const float4v*)(ilb + p));
    float4v c0 = *(const float4v*)(isb + p);
    float4v c1 = *(const float4v*)(isb + p + HW);
    float4v c2 = *(const float4v*)(isb + p + 2 * HW);
    float4v c3 = *(const float4v*)(isb + p + 3 * HW);
#pragma unroll
    for (int k = 0; k < 4; ++k) {
      int lab = lab4[k];
      float m0 = 0.0f, m1 = 0.0f, m2 = 0.0f, m3 = 0.0f;
      if (lab > 0) {
        m0 = s_mean[lab * CH + 0];
        m1 = s_mean[lab * CH + 1];
        m2 = s_mean[lab * CH + 2];
        m3 = s_mean[lab * CH + 3];
      }
      float d0 = c0[k] - m0, d1 = c1[k] - m1, d2 = c2[k] - m2, d3 = c3[k] - m3;
      float norm = d0 * d0 + d1 * d1 + d2 * d2 + d3 * d3;
      if (il4[k] > LINE_THRESH) acc += norm;
    }
  }

  // wave32 reduction
  for (int off = 16; off > 0; off >>= 1) acc += __shfl_down(acc, off, 32);
  const int wave = tid >> 5, lane = tid & 31;
  if (lane == 0) s_red[wave] = acc;
  __syncthreads();
  if (wave == 0) {
    float v = (lane < (TPB / 32)) ? s_red[lane] : 0.0f;
    for (int off = (TPB / 64); off > 0; off >>= 1) v += __shfl_down(v, off, 32);
    if (lane == 0) {
      const float invN = 1.0f / ((float)BATCH * (float)HW);
      atomicAdd(out, v * invN);
    }
  }
}

extern "C" void kernel_launch(void* const* d_in, const int* in_sizes, int n_in,
                              void* d_out, int out_size, void* d_ws, size_t ws_size,
                              hipStream_t stream) {
  (void)in_sizes; (void)n_in; (void)out_size; (void)ws_size;
  const float* Is = (const float*)d_in[0];
  const int* Ispp = (const int*)d_in[1];
  const float* Il = (const float*)d_in[2];
  float* out = (float*)d_out;

  float* ws = (float*)d_ws;
  float* sums = ws;                                      // B*1025*4
  float* cnts = sums + (size_t)BATCH * NUM_LABELS * CH;  // B*1025
  float* means = cnts + (size_t)BATCH * NUM_LABELS;      // B*1025*4

  const int initN = BATCH * NUM_LABELS * (CH + 1);  // sums + cnts are contiguous
  init_kernel<<<(initN + TPB - 1) / TPB, TPB, 0, stream>>>(sums, initN, out);
  pass1_segsum<<<BATCH * BLOCKS1_PER_IMG, TPB, 0, stream>>>(Is, Ispp, sums, cnts);
  means_kernel<<<(BATCH * NUM_LABELS + TPB - 1) / TPB, TPB, 0, stream>>>(sums, cnts, means);
  pass2_loss<<<BATCH * BLOCKS2_PER_IMG, TPB, 0, stream>>>(Is, Ispp, Il, means, out);
}